// SimpleGCN_34900904247990
// MI455X (gfx1250) — compile-verified
//
#include <hip/hip_runtime.h>

typedef __attribute__((ext_vector_type(2))) float v2f;
typedef __attribute__((ext_vector_type(8))) float v8f;

// ---------------------------------------------------------------------------
// deg[i] = 1.0  (self-loop contribution)
__global__ void gcn_set1(float* __restrict__ p, int n) {
  int i = blockIdx.x * 256 + threadIdx.x;
  if (i < n) p[i] = 1.0f;
}

// deg[dst[e]] += 1  over all edges (float count is exact for deg < 2^24)
__global__ void gcn_count_deg(const long long* __restrict__ dst,
                              float* __restrict__ deg, int E) {
  int i = blockIdx.x * 256 + threadIdx.x;
  if (i < E) unsafeAtomicAdd(deg + dst[i], 1.0f);
}

// dinv = rsqrt(deg)   (deg >= 1 always, no zero guard needed)
__global__ void gcn_rsqrt(float* __restrict__ p, int n) {
  int i = blockIdx.x * 256 + threadIdx.x;
  if (i < n) p[i] = rsqrtf(p[i]);
}

// ---------------------------------------------------------------------------
// Fused GCN-layer GEMM via V_WMMA_F32_16X16X4_F32 (fp32 matrix pipe):
//   H[M,64] = act(A)[M,K] @ W[K,64]          (raw product, edge-gather source)
//   O[M,64] = bias + dinv^2 * H              (self-loop-initialized accumulator)
// act = ReLU when RELU_IN (layer 2 input), identity otherwise.
//
// Block = 128 threads = 4 waves; block b covers rows [16b,16b+16), wave w owns
// column tile [16w,16w+16). M must be a multiple of 16 (100000 = 16*6250).
//
// fp32 WMMA VGPR layouts (ISA 7.12.2):
//   A 16x4:  lanes 0-15 hold M=lane,  {v0,v1}={K0,K1}; lanes 16-31: {K2,K3}
//   B 4x16:  lanes 0-15 hold N=lane,  {v0,v1}={K0,K1}; lanes 16-31: {K2,K3}
//   C/D 16x16: VGPR g -> M=g (lanes 0-15) / M=8+g (lanes 16-31), N=lane%16
template <int K, bool RELU_IN>
__global__ void __launch_bounds__(128)
gcn_gemm_wmma_fused(const float* __restrict__ A, const float* __restrict__ W,
                    const float* __restrict__ dinv, const float* __restrict__ bias,
                    float* __restrict__ H, float* __restrict__ O) {
  const int lane    = threadIdx.x & 31;
  const int wave    = threadIdx.x >> 5;
  const int rowBase = blockIdx.x * 16;
  const int colBase = wave * 16;
  const int hi      = lane >> 4;   // half-wave select
  const int l16     = lane & 15;
  const int kOff    = hi * 2;

  const float* aRow = A + (size_t)(rowBase + l16) * K;
  const float  bval = bias[colBase + l16];

  v8f acc = {};
#pragma unroll
  for (int k = 0; k < K; k += 4) {
    v2f a = *(const v2f*)(aRow + k + kOff);                 // 8B coalesced
    if (RELU_IN) {
      a.x = fmaxf(a.x, 0.0f);
      a.y = fmaxf(a.y, 0.0f);
    }
    v2f b;
    b.x = W[(size_t)(k + kOff) * 64 + colBase + l16];
    b.y = W[(size_t)(k + kOff + 1) * 64 + colBase + l16];
    acc = __builtin_amdgcn_wmma_f32_16x16x4_f32(
        /*neg_a=*/false, a, /*neg_b=*/false, b,
        /*c_mod=*/(short)0, acc, /*reuse_a=*/false, /*reuse_b=*/false);
  }

  // Fused epilogue: write raw product + self-loop/bias init, straight from VGPRs.
#pragma unroll
  for (int g = 0; g < 8; ++g) {
    const int row = rowBase + hi * 8 + g;
    const float dv = dinv[row];           // broadcast load across 16 lanes
    const float hv = acc[g];
    H[(size_t)row * 64 + colBase + l16] = hv;
    O[(size_t)row * 64 + colBase + l16] = bval + dv * dv * hv;
  }
}

// ---------------------------------------------------------------------------
// For each edge (s,d): O[d,:] += dinv[s]*dinv[d] * H[s,:]
// One wave32 per edge; each lane owns a float2 -> one wave covers the whole
// 256B feature row (cacheline-granular gather + coalesced fp32 atomics).
__global__ void __launch_bounds__(256)
gcn_scatter(const long long* __restrict__ src, const long long* __restrict__ dst,
            const float* __restrict__ dinv, const float* __restrict__ H,
            float* __restrict__ O, int E) {
  int e = blockIdx.x * 8 + (threadIdx.x >> 5);
  if (e >= E) return;
  int lane = threadIdx.x & 31;
  long long s = src[e];
  long long d = dst[e];
  float coef = dinv[s] * dinv[d];
  float2 hv = ((const float2*)(H + (size_t)s * 64))[lane];
  float* op = O + (size_t)d * 64 + lane * 2;
  unsafeAtomicAdd(op,     coef * hv.x);   // -> global_atomic_add_f32
  unsafeAtomicAdd(op + 1, coef * hv.y);
}

// ---------------------------------------------------------------------------
extern "C" void kernel_launch(void* const* d_in, const int* in_sizes, int n_in,
                              void* d_out, int out_size, void* d_ws, size_t ws_size,
                              hipStream_t stream) {
  const float*     x  = (const float*)d_in[0];
  const long long* ei = (const long long*)d_in[1];   // int64 edge_index [2,E]
  const float*     W1 = (const float*)d_in[2];
  const float*     b1 = (const float*)d_in[3];
  const float*     W2 = (const float*)d_in[4];
  const float*     b2 = (const float*)d_in[5];

  const int N = in_sizes[0] / 128;   // 100000 nodes
  const int E = in_sizes[1] / 2;     // 1000000 edges
  const long long* src = ei;
  const long long* dst = ei + E;

  // workspace: dinv[N] | h[N*64] | out1[N*64]  (~52 MB)
  float* dinv = (float*)d_ws;
  float* h    = dinv + (((size_t)N + 63) & ~(size_t)63);
  float* out1 = h + (size_t)N * 64;
  float* out  = (float*)d_out;

  const dim3 blk(256);

  // --- normalization coefficients ---
  gcn_set1     <<<dim3((N + 255) / 256), blk, 0, stream>>>(dinv, N);
  gcn_count_deg<<<dim3((E + 255) / 256), blk, 0, stream>>>(dst, dinv, E);
  gcn_rsqrt    <<<dim3((N + 255) / 256), blk, 0, stream>>>(dinv, N);

  // --- layer 1: h = x@W1 ; out1 = b1 + dinv^2*h ; then edge scatter-add ---
  gcn_gemm_wmma_fused<128, false>
      <<<dim3(N / 16), dim3(128), 0, stream>>>(x, W1, dinv, b1, h, out1);
  gcn_scatter<<<dim3((E + 7) / 8), blk, 0, stream>>>(src, dst, dinv, h, out1, E);

  // --- layer 2: g = relu(out1)@W2 ; out = b2 + dinv^2*g ; edge scatter-add ---
  gcn_gemm_wmma_fused<64, true>
      <<<dim3(N / 16), dim3(128), 0, stream>>>(out1, W2, dinv, b2, h, out);
  gcn_scatter<<<dim3((E + 7) / 8), blk, 0, stream>>>(src, dst, dinv, h, out, E);
}